// LocalDensityEncoder_76922864271780
// MI455X (gfx1250) — compile-verified
//
#include <hip/hip_runtime.h>
#include <math.h>
#include <stdint.h>

// Problem constants (match reference setup_inputs)
#define N_NODES 16384
#define N_EDGES 524288
#define HDIM    256
#define HID     128
#define RADIUS2 2500.0f
#define EPSV    1e-8f

typedef float v2f __attribute__((ext_vector_type(2)));
typedef float v8f __attribute__((ext_vector_type(8)));
typedef unsigned int u32x4 __attribute__((ext_vector_type(4)));
typedef int i32x4 __attribute__((ext_vector_type(4)));
typedef int i32x8 __attribute__((ext_vector_type(8)));

__device__ __forceinline__ v8f wmma_f32_16x16x4(v2f a, v2f b, v8f c) {
  // D = A(16x4,f32) x B(4x16,f32) + C(16x16,f32)
  return __builtin_amdgcn_wmma_f32_16x16x4_f32(
      /*neg_a=*/false, a, /*neg_b=*/false, b,
      /*c_mod=*/(short)0, c, /*reuse_a=*/false, /*reuse_b=*/false);
}

// TDM 1-D contiguous copy: nfloats f32 elements global -> LDS.
// D# per CDNA5 ISA ch.8: group0 {count=1, lds_addr, global_addr, type=2},
// group1 {data_size=4B, tensor_dim0=tile_dim0=nfloats, dims1=1, stride=nfloats}.
__device__ __forceinline__ void tdm_load_1d(unsigned lds_byte_addr,
                                            const void* gptr, unsigned nfloats) {
  unsigned long long ga = (unsigned long long)(uintptr_t)gptr;
  u32x4 g0;
  g0[0] = 1u;                                             // count=1 (user mode)
  g0[1] = lds_byte_addr;                                  // lds_addr
  g0[2] = (unsigned)(ga & 0xffffffffu);                   // global_addr[31:0]
  g0[3] = (unsigned)((ga >> 32) & 0x01ffffffu) | (2u << 30); // addr[56:32] | type=2
  i32x8 g1;
  g1[0] = (2 << 16);                                      // data_size = 4 bytes
  g1[1] = (int)((nfloats & 0xffffu) << 16);               // tensor_dim0[15:0]
  g1[2] = (int)((nfloats >> 16) | (1u << 16));            // tensor_dim0[31:16] | tensor_dim1=1
  g1[3] = (int)((nfloats & 0xffffu) << 16);               // tile_dim0
  g1[4] = 1;                                              // tile_dim1=1, tile_dim2=0
  g1[5] = (int)nfloats;                                   // tensor_dim0_stride[31:0]
  g1[6] = 0;                                              // stride hi / dim1_stride lo
  g1[7] = 0;
  i32x4 z4 = {0, 0, 0, 0};
  i32x8 z8 = {0, 0, 0, 0, 0, 0, 0, 0};
  __builtin_amdgcn_tensor_load_to_lds(g0, g1, z4, z4, z8, 0);
}

// ---------------- utility kernels ----------------

__global__ void k_zero(float* __restrict__ p, int n) {
  int i = blockIdx.x * blockDim.x + threadIdx.x;
  if (i < n) p[i] = 0.0f;
}

// Per-point descriptors so pairwise dist^2 becomes a K=4 matmul:
//   A-row(i) = ( x_i, y_i, sq_i, 1 ),  B-col(j) = (-2x_j,-2y_j, 1, sq_j )
//   A.B = sq_i + sq_j - 2(x_i x_j + y_i y_j) = d2(i,j)
__global__ void k_prep(const float* __restrict__ coords,
                       float* __restrict__ Am, float* __restrict__ Bm) {
  int i = blockIdx.x * blockDim.x + threadIdx.x;
  if (i < N_NODES) {
    float x = coords[2 * i], y = coords[2 * i + 1];
    float sq = x * x + y * y;
    Am[4 * i + 0] = x;        Am[4 * i + 1] = y;
    Am[4 * i + 2] = sq;       Am[4 * i + 3] = 1.0f;
    Bm[4 * i + 0] = -2.f * x; Bm[4 * i + 1] = -2.f * y;
    Bm[4 * i + 2] = 1.0f;     Bm[4 * i + 3] = sq;
  }
}

__global__ void k_degree(const int* __restrict__ row, float* __restrict__ degree) {
  int e = blockIdx.x * blockDim.x + threadIdx.x;
  if (e < N_EDGES) atomicAdd(&degree[row[e]], 1.0f);
}

// scatter x[col[e]] into nbr_sum[row[e]]; 64 threads/edge, float4 per thread
__global__ void k_scatter(const int* __restrict__ row, const int* __restrict__ col,
                          const float* __restrict__ x, float* __restrict__ nbr) {
  int tid = blockIdx.x * blockDim.x + threadIdx.x;     // < E * 64
  int e = tid >> 6;
  int c = (tid & 63) * 4;
  int r = row[e], cl = col[e];
  const float4 v = *(const float4*)(x + cl * HDIM + c);
  float* dst = nbr + r * HDIM + c;
  atomicAdd(dst + 0, v.x);
  atomicAdd(dst + 1, v.y);
  atomicAdd(dst + 2, v.z);
  atomicAdd(dst + 3, v.w);
}

// ---------------- WMMA pairwise density with TDM-staged B ----------------
// 8 waves/block, each wave owns a 16-row i-strip. Wave 0 DMA's 16KB chunks of
// Bm into a double-buffered LDS staging area with tensor_load_to_lds; all 8
// waves consume B fragments from LDS (ds_load_b64), one WMMA per 16x16 tile.
#define JCHUNK 1024                       // j-columns per chunk
#define CHUNK_FLOATS (JCHUNK * 4)         // 4096 floats = 16KB

__global__ void k_density(const float* __restrict__ Am, const float* __restrict__ Bm,
                          float* __restrict__ density) {
  __shared__ float bbuf[2][CHUNK_FLOATS]; // 32KB
  int lane = threadIdx.x & 31;
  int wave = threadIdx.x >> 5;
  int strip = blockIdx.x * 8 + wave;
  int i0 = strip * 16;
  int half = lane >> 4;                   // 0: K={0,1}, 1: K={2,3}
  int l15 = lane & 15;

  const float2* Ap = (const float2*)Am;
  float2 af = Ap[(i0 + l15) * 2 + half];
  v2f a; a.x = af.x; a.y = af.y;

  float cnt[8];
#pragma unroll
  for (int g = 0; g < 8; ++g) cnt[g] = 0.0f;

  unsigned lds0 = (unsigned)(uintptr_t)&bbuf[0][0];
  unsigned lds1 = (unsigned)(uintptr_t)&bbuf[1][0];

  const int nchunks = N_NODES / JCHUNK;   // 16
  if (wave == 0)
    tdm_load_1d(lds0, Bm, CHUNK_FLOATS);

  for (int c = 0; c < nchunks; ++c) {
    if (wave == 0) {
      if (c + 1 < nchunks) {
        tdm_load_1d(((c + 1) & 1) ? lds1 : lds0,
                    Bm + (size_t)(c + 1) * CHUNK_FLOATS, CHUNK_FLOATS);
        __builtin_amdgcn_s_wait_tensorcnt(1);   // chunk c has landed
      } else {
        __builtin_amdgcn_s_wait_tensorcnt(0);
      }
    }
    __syncthreads();                            // chunk c visible to all waves

    const float2* Bp = (const float2*)((c & 1) ? &bbuf[1][0] : &bbuf[0][0]);
    for (int jt = 0; jt < JCHUNK / 16; ++jt) {  // 64 tiles per chunk
      float2 bf = Bp[(jt * 16 + l15) * 2 + half];
      v2f b; b.x = bf.x; b.y = bf.y;
      v8f cc = {};
      v8f d = wmma_f32_16x16x4(a, b, cc);       // d[g]=d2(i0+g+8*half, j)
#pragma unroll
      for (int g = 0; g < 8; ++g) cnt[g] += (d[g] <= RADIUS2) ? 1.0f : 0.0f;
    }
    __syncthreads();                            // done reading before next DMA reuses buffer
  }

  // reduce over the 16 lanes of each half (N dimension)
#pragma unroll
  for (int g = 0; g < 8; ++g) {
    float v = cnt[g];
    v += __shfl_xor(v, 1, 32);
    v += __shfl_xor(v, 2, 32);
    v += __shfl_xor(v, 4, 32);
    v += __shfl_xor(v, 8, 32);
    cnt[g] = v;
  }
  if (l15 == 0) {
#pragma unroll
    for (int g = 0; g < 8; ++g)
      density[i0 + g + 8 * half] = cnt[g] - 1.0f;   // exclude self
  }
}

// ---------------- per-node feature variance ----------------
__global__ void k_fvar(const float* __restrict__ x, const float* __restrict__ nbr,
                       const float* __restrict__ degree, float* __restrict__ fvar) {
  __shared__ float red[256];
  int n = blockIdx.x;
  int t = threadIdx.x;
  float deg = degree[n];
  float cnt = deg > 1.0f ? deg : 1.0f;
  float diff = x[n * HDIM + t] - nbr[n * HDIM + t] / cnt;
  red[t] = diff * diff;
  __syncthreads();
#pragma unroll
  for (int s = 128; s > 0; s >>= 1) {
    if (t < s) red[t] += red[t + s];
    __syncthreads();
  }
  if (t == 0) fvar[n] = sqrtf(red[0]);
}

// maxima of three non-negative float arrays via uint-bit atomicMax
__global__ void k_max3(const float* __restrict__ degree, const float* __restrict__ density,
                       const float* __restrict__ fvar, unsigned* __restrict__ mx) {
  int i = blockIdx.x * blockDim.x + threadIdx.x;
  if (i < N_NODES) {
    atomicMax(&mx[0], __float_as_uint(degree[i]));
    atomicMax(&mx[1], __float_as_uint(density[i]));
    atomicMax(&mx[2], __float_as_uint(fvar[i]));
  }
}

// ---------------- WMMA MLP: relu(feats@w1+b1)@w2+b2 ----------------
__global__ void k_mlp(const float* __restrict__ degree, const float* __restrict__ density,
                      const float* __restrict__ fvar, const unsigned* __restrict__ mx,
                      const float* __restrict__ w1, const float* __restrict__ b1,
                      const float* __restrict__ w2, const float* __restrict__ b2,
                      float* __restrict__ out) {
  __shared__ float h1s[4][16 * HID];   // 32 KB
  int lane = threadIdx.x & 31;
  int wave = threadIdx.x >> 5;
  int half = lane >> 4;
  int l15 = lane & 15;
  int i0 = (blockIdx.x * 4 + wave) * 16;

  float maxdeg = __uint_as_float(mx[0]) + EPSV;
  float maxden = __uint_as_float(mx[1]) + EPSV;
  float maxfv  = __uint_as_float(mx[2]) + EPSV;

  // A fragment of feats (16x3 padded to 16x4)
  int r = i0 + l15;
  v2f a1;
  if (half == 0) { a1.x = degree[r] / maxdeg; a1.y = density[r] / maxden; }
  else           { a1.x = fvar[r] / maxfv;    a1.y = 0.0f; }

  // layer 1: 8 N-tiles of 16, K=4 (3 real + zero pad)
  for (int n = 0; n < HID / 16; ++n) {
    int colc = n * 16 + l15;
    v2f b;
    if (half == 0) { b.x = w1[0 * HID + colc]; b.y = w1[1 * HID + colc]; }
    else           { b.x = w1[2 * HID + colc]; b.y = 0.0f; }
    float bb = b1[colc];
    v8f c;
#pragma unroll
    for (int g = 0; g < 8; ++g) c[g] = bb;
    v8f d = wmma_f32_16x16x4(a1, b, c);
#pragma unroll
    for (int g = 0; g < 8; ++g) {
      float vv = d[g] > 0.0f ? d[g] : 0.0f;     // relu
      h1s[wave][(g + 8 * half) * HID + colc] = vv;
    }
  }
  __syncthreads();

  // layer 2: 16 N-tiles of 16, K=128 as 32 chained WMMAs
  for (int n = 0; n < HDIM / 16; ++n) {
    int colc = n * 16 + l15;
    float bb = b2[colc];
    v8f acc;
#pragma unroll
    for (int g = 0; g < 8; ++g) acc[g] = bb;
    for (int kk = 0; kk < HID / 4; ++kk) {
      int kb = kk * 4 + 2 * half;   // lanes<16: K={4k,4k+1}; lanes>=16: K={4k+2,4k+3}
      v2f a; a.x = h1s[wave][l15 * HID + kb];
             a.y = h1s[wave][l15 * HID + kb + 1];
      v2f b; b.x = w2[(kb)     * HDIM + colc];
             b.y = w2[(kb + 1) * HDIM + colc];
      acc = wmma_f32_16x16x4(a, b, acc);
    }
#pragma unroll
    for (int g = 0; g < 8; ++g)
      out[(i0 + g + 8 * half) * HDIM + colc] = acc[g];
  }
}

// ---------------- launch ----------------

extern "C" void kernel_launch(void* const* d_in, const int* in_sizes, int n_in,
                              void* d_out, int out_size, void* d_ws, size_t ws_size,
                              hipStream_t stream) {
  const float* x      = (const float*)d_in[0];
  const int*   ei     = (const int*)  d_in[1];
  const float* coords = (const float*)d_in[2];
  const float* w1     = (const float*)d_in[3];
  const float* b1     = (const float*)d_in[4];
  const float* w2     = (const float*)d_in[5];
  const float* b2     = (const float*)d_in[6];
  float* out = (float*)d_out;

  const int* row = ei;
  const int* col = ei + N_EDGES;

  // workspace layout (floats): degree | nbr_sum | maxima(16) | density | fvar | Am | Bm
  float* ws      = (float*)d_ws;
  float* degree  = ws;
  float* nbr     = degree + N_NODES;
  unsigned* mx   = (unsigned*)(nbr + (size_t)N_NODES * HDIM);
  float* density = (float*)(mx + 16);
  float* fvar    = density + N_NODES;
  float* Am      = fvar + N_NODES;
  float* Bm      = Am + 4 * N_NODES;

  // zero degree + nbr_sum + maxima (contiguous prefix)
  int zn = N_NODES + N_NODES * HDIM + 16;
  k_zero<<<(zn + 255) / 256, 256, 0, stream>>>(ws, zn);

  k_prep<<<(N_NODES + 255) / 256, 256, 0, stream>>>(coords, Am, Bm);

  k_degree<<<(N_EDGES + 255) / 256, 256, 0, stream>>>(row, degree);

  k_scatter<<<(N_EDGES * (HDIM / 4)) / 256, 256, 0, stream>>>(row, col, x, nbr);

  // 1024 strips of 16 rows, 8 waves/block
  k_density<<<(N_NODES / 16) / 8, 256, 0, stream>>>(Am, Bm, density);

  k_fvar<<<N_NODES, 256, 0, stream>>>(x, nbr, degree, fvar);

  k_max3<<<(N_NODES + 255) / 256, 256, 0, stream>>>(degree, density, fvar, mx);

  // 64 rows/block (4 waves x 16 rows)
  k_mlp<<<N_NODES / 64, 128, 0, stream>>>(degree, density, fvar, mx, w1, b1, w2, b2, out);
}